// GetV_18983755448816
// MI455X (gfx1250) — compile-verified
//
#include <hip/hip_runtime.h>

// SO(3) left-Jacobian V = I + B*[w]x + C*[w]x^2 for 4M rows.
// Bandwidth-bound: 192 MB total traffic -> ~8.2 us floor @ 23.3 TB/s.
// Strategy: LDS-staged, fully coalesced b128 traffic using CDNA5 async
// global<->LDS copies (ASYNCcnt), compute in registers, v_sin/v_cos HW trans.

#define TILE    256
#define THREADS 256
#define EPSF    1e-10f

// Builtin parameter type (from clang diagnostic): v4i32 with HIP language
// address spaces: __device__ == addrspace(1), __shared__ == addrspace(3).
typedef int v4i __attribute__((vector_size(4 * sizeof(int))));
typedef __attribute__((address_space(1))) v4i gv4i;   // global
typedef __attribute__((address_space(3))) v4i lv4i;   // LDS

__device__ __forceinline__ void async_load_b128(const float4* g, float4* l) {
#if __has_builtin(__builtin_amdgcn_global_load_async_to_lds_b128)
  __builtin_amdgcn_global_load_async_to_lds_b128(
      (gv4i*)g, (lv4i*)l, /*offset=*/0, /*cpol=*/0);
#else
  unsigned lds_off = (unsigned)(unsigned long long)(const void*)l;
  asm volatile("global_load_async_to_lds_b128 %0, %1, off"
               :
               : "v"(lds_off), "v"(g)
               : "memory");
#endif
}

__device__ __forceinline__ void async_store_b128(float4* g, const float4* l) {
#if __has_builtin(__builtin_amdgcn_global_store_async_from_lds_b128)
  __builtin_amdgcn_global_store_async_from_lds_b128(
      (gv4i*)g, (lv4i*)l, /*offset=*/0, /*cpol=*/0);
#else
  unsigned lds_off = (unsigned)(unsigned long long)(const void*)l;
  asm volatile("global_store_async_from_lds_b128 %0, %1, off"
               :
               : "v"(g), "v"(lds_off)
               : "memory");
#endif
}

__device__ __forceinline__ void wait_asynccnt0() {
#if __has_builtin(__builtin_amdgcn_s_wait_asynccnt)
  __builtin_amdgcn_s_wait_asynccnt(0);
#else
  asm volatile("s_wait_asynccnt 0" ::: "memory");
#endif
}

__device__ __forceinline__ void compute_row(float x, float y, float z,
                                            float* __restrict__ o) {
  x += EPSF; y += EPSF; z += EPSF;
  const float xx = x * x, yy = y * y, zz = z * z;
  const float t2 = xx + yy + zz;
  const float t  = __builtin_sqrtf(t2);
  const float s  = __sinf(t);          // v_sin_f32 (HW transcendental)
  const float c  = __cosf(t);          // v_cos_f32
  const float inv_t2 = 1.0f / t2;
  const float A = s / t;               // sin(t)/t
  const float B = (1.0f - c) * inv_t2; // (1-cos)/t^2
  const float C = (1.0f - A) * inv_t2; // (1-A)/t^2
  const float xy = x * y, xz = x * z, yz = y * z;
  // skew^2 = w w^T - t^2 I  (matches reference skew@skew algebraically)
  o[0] = 1.0f - C * (yy + zz);
  o[1] = C * xy - B * z;
  o[2] = C * xz + B * y;
  o[3] = C * xy + B * z;
  o[4] = 1.0f - C * (xx + zz);
  o[5] = C * yz - B * x;
  o[6] = C * xz - B * y;
  o[7] = C * yz + B * x;
  o[8] = 1.0f - C * (xx + yy);
}

__global__ __launch_bounds__(THREADS) void getv_kernel(
    const float* __restrict__ in, float* __restrict__ out, int nrows) {
  __shared__ alignas(16) float lds_in [TILE * 3];  // 3 KB
  __shared__ alignas(16) float lds_out[TILE * 9];  // 9 KB

  const int tid = threadIdx.x;
  const long long rowBase = (long long)blockIdx.x * TILE;

  if (rowBase + TILE <= (long long)nrows) {
    // ---- full tile: async-staged, fully coalesced b128 path ----
    // Stage 256 rows (192 float4) into LDS: waves 0..5 issue one b128 each.
    const float4* g4  = (const float4*)(in + rowBase * 3);
    float4*       li4 = (float4*)lds_in;
    if (tid < (TILE * 3 / 4)) {
      async_load_b128(g4 + tid, li4 + tid);
    }
    wait_asynccnt0();     // each wave waits for its own async loads
    __syncthreads();      // ... then all waves' data is visible

    // Per-row compute from LDS (stride-3 word reads: conflict-free, 3⊥64)
    compute_row(lds_in[tid * 3 + 0], lds_in[tid * 3 + 1], lds_in[tid * 3 + 2],
                &lds_out[tid * 9]);  // stride-9 word writes: conflict-free

    __syncthreads();      // all 2304 outputs resident in LDS

    // Drain tile: 576 float4 async stores (base = 9216*blockIdx bytes, 16B-aligned)
    float4*       o4  = (float4*)(out + rowBase * 9);
    const float4* lo4 = (const float4*)lds_out;
    async_store_b128(o4 + tid,       lo4 + tid);
    async_store_b128(o4 + tid + 256, lo4 + tid + 256);
    if (tid < 64) {       // waves 0,1 only — no intra-wave divergence
      async_store_b128(o4 + tid + 512, lo4 + tid + 512);
    }
    wait_asynccnt0();     // (s_endpgm also implies wait-idle)
  } else {
    // ---- tail tile (unused for N=4M, kept for generality) ----
    const long long r = rowBase + tid;
    if (r < (long long)nrows) {
      float o[9];
      compute_row(in[r * 3 + 0], in[r * 3 + 1], in[r * 3 + 2], o);
#pragma unroll
      for (int j = 0; j < 9; ++j) out[r * 9 + j] = o[j];
    }
  }
}

extern "C" void kernel_launch(void* const* d_in, const int* in_sizes, int n_in,
                              void* d_out, int out_size, void* d_ws, size_t ws_size,
                              hipStream_t stream) {
  (void)n_in; (void)out_size; (void)d_ws; (void)ws_size;
  const float* dw  = (const float*)d_in[0];
  float*       out = (float*)d_out;
  const int nrows  = in_sizes[0] / 3;               // 4,000,000
  const int blocks = (nrows + TILE - 1) / TILE;     // 15,625 (exact tiles)
  getv_kernel<<<blocks, THREADS, 0, stream>>>(dw, out, nrows);
}